// GlobalNodeAttentionHTR_50208167690487
// MI455X (gfx1250) — compile-verified
//
#include <hip/hip_runtime.h>
#include <hip/hip_bf16.h>
#include <math.h>

#define N_NODES 1024
#define C_CH    128
#define H_HEADS 8
#define D_HEAD  16
#define NM      9          // total SH components (1+3+5)
#define ROWLEN  (NM * C_CH)
#define KC      64         // K-chunk staged in LDS for big GEMMs
#define APAD    68         // padded LDS row stride (68*4B = 272B: 16B-aligned, bank-spread)

typedef __attribute__((ext_vector_type(2))) float v2f;
typedef __attribute__((ext_vector_type(8))) float v8f;
typedef __attribute__((ext_vector_type(4))) int   v4i;

#if defined(__HIP_DEVICE_COMPILE__) && __has_builtin(__builtin_amdgcn_global_load_async_to_lds_b128) && __has_builtin(__builtin_amdgcn_s_wait_asynccnt)
#define HAS_ASYNC_LDS 1
#else
#define HAS_ASYNC_LDS 0
#endif

typedef __attribute__((address_space(1))) v4i* g_v4i_p;
typedef __attribute__((address_space(3))) v4i* l_v4i_p;

// fp32 WMMA: D(16x16) = A(16x4) * B(4x16) + C   (V_WMMA_F32_16X16X4_F32)
__device__ __forceinline__ v8f wmma4(v2f a, v2f b, v8f c) {
  return __builtin_amdgcn_wmma_f32_16x16x4_f32(false, a, false, b, (short)0, c,
                                               false, false);
}

__device__ __forceinline__ int l_of_m(int m) { return (m == 0) ? 0 : ((m < 4) ? 1 : 2); }

// Stage a 16 x KC fp32 A-tile (row stride `stride` floats) into LDS.
// 8 x B128 per-lane async transfers: instr t covers rows 2t,2t+1; each lane moves 16B.
__device__ __forceinline__ void stage_a_tile(const float* __restrict__ gbase,
                                             size_t stride, int k0,
                                             float* __restrict__ lds, int lane) {
  int r  = lane >> 4;         // 0..1
  int cg = (lane & 15) * 4;   // 0,4,...,60
#pragma unroll
  for (int t = 0; t < 8; ++t) {
    int row = 2 * t + r;
    const float* g = gbase + (size_t)row * stride + k0 + cg;
    float* l = lds + row * APAD + cg;
#if HAS_ASYNC_LDS
    __builtin_amdgcn_global_load_async_to_lds_b128((g_v4i_p)g, (l_v4i_p)l, 0, 0);
#else
    *(float4*)l = *(const float4*)g;
#endif
  }
}

__device__ __forceinline__ void wait_async_le(int n) {
#if HAS_ASYNC_LDS
  if (n == 0)
    __builtin_amdgcn_s_wait_asynccnt(0);
  else
    __builtin_amdgcn_s_wait_asynccnt(8);
#else
  (void)n;
#endif
}

// ---------------------------------------------------------------------------
// K1: per-pair SH weights (already / (2l+1), masked). Layout SHW[m][j][i].
// ---------------------------------------------------------------------------
__global__ void k_pair_sh(const float* __restrict__ pos, const int* __restrict__ batch,
                          float* __restrict__ SHW) {
  int idx = blockIdx.x * blockDim.x + threadIdx.x;   // idx = j*N + i
  int j = idx >> 10;
  int i = idx & (N_NODES - 1);
  float dx = pos[i * 3 + 0] - pos[j * 3 + 0];
  float dy = pos[i * 3 + 1] - pos[j * 3 + 1];
  float dz = pos[i * 3 + 2] - pos[j * 3 + 2];
  float d = fmaxf(sqrtf(dx * dx + dy * dy + dz * dz), 1e-8f);
  float x = dx / d, y = dy / d, z = dz / d;
  float vm = ((batch[i] == batch[j]) && (i != j)) ? 1.f : 0.f;
  const float INV = 0.28209479177387814f;           // 1/sqrt(4pi)
  const float C1  = 0.4886025119029199f;            // sqrt(3/(4pi))
  const float S15 = 3.872983346207417f * INV;       // sqrt(15)/sqrt(4pi)
  const float S5  = 2.23606797749979f * INV;        // sqrt(5)/sqrt(4pi)
  float w[9];
  w[0] = INV;
  w[1] = C1 * x * (1.f / 3.f);
  w[2] = C1 * y * (1.f / 3.f);
  w[3] = C1 * z * (1.f / 3.f);
  w[4] = S15 * x * z * 0.2f;
  w[5] = S15 * x * y * 0.2f;
  w[6] = S5 * (y * y - 0.5f * (x * x + z * z)) * 0.2f;
  w[7] = S15 * y * z * 0.2f;
  w[8] = 0.5f * S15 * (z * z - x * x) * 0.2f;
  size_t base = (size_t)j * N_NODES + i;
#pragma unroll
  for (int m = 0; m < 9; ++m)
    SHW[(size_t)m * N_NODES * N_NODES + base] = w[m] * vm;
}

// ---------------------------------------------------------------------------
// K2: W_row[i,m] = (1/N) sum_j SHW[m][j][i]
// ---------------------------------------------------------------------------
__global__ void k_rowsum(const float* __restrict__ SHW, float* __restrict__ Wrow) {
  int t = blockIdx.x * blockDim.x + threadIdx.x;    // t = m*N + i
  int m = t >> 10;
  int i = t & (N_NODES - 1);
  if (m >= NM) return;
  const float* p = SHW + (size_t)m * N_NODES * N_NODES + i;
  float s = 0.f;
  for (int j = 0; j < N_NODES; ++j) s += p[(size_t)j * N_NODES];
  Wrow[i * NM + m] = s * (1.f / N_NODES);
}

// ---------------------------------------------------------------------------
// K3: row-mean[i,c] = sum_m f[i,m,c] * W_row[i,m]
// ---------------------------------------------------------------------------
__global__ void k_rowmean(const float* __restrict__ x_emb, const float* __restrict__ Wrow,
                          float* __restrict__ rm) {
  int t = blockIdx.x * blockDim.x + threadIdx.x;    // t = i*C + c
  int i = t >> 7;
  int c = t & (C_CH - 1);
  float s = 0.f;
#pragma unroll
  for (int m = 0; m < 9; ++m)
    s += x_emb[(size_t)i * ROWLEN + m * C_CH + c] * Wrow[i * NM + m];
  rm[t] = s;
}

// ---------------------------------------------------------------------------
// K4: col-mean[j,c] = (1/N) sum_m sum_i SHW[m][j][i] * f[i,m,c]   (WMMA GEMM)
// one wave per 16x16 tile; A chunks async-staged to LDS, double-buffered.
// ---------------------------------------------------------------------------
__global__ void k_colmean(const float* __restrict__ SHW, const float* __restrict__ x_emb,
                          float* __restrict__ cm) {
  int jt = blockIdx.x >> 3, ct = blockIdx.x & 7;
  int j0 = jt * 16, c0 = ct * 16;
  int lane = threadIdx.x & 31;
  int lhalf = lane >> 4;
  int arow = lane & 15, bcol = lane & 15;
  int kh = lhalf * 2;
  __shared__ __align__(16) float As[2][16 * APAD];
  v8f acc = {};
  for (int m = 0; m < 9; ++m) {
    const float* A = SHW + (size_t)m * N_NODES * N_NODES + (size_t)j0 * N_NODES;
    const float* B = x_emb + m * C_CH;
    stage_a_tile(A, N_NODES, 0, As[0], lane);
    for (int kc = 0; kc < N_NODES / KC; ++kc) {
      int k0 = kc * KC;
      if (kc + 1 < N_NODES / KC) {
        stage_a_tile(A, N_NODES, k0 + KC, As[(kc + 1) & 1], lane);
        wait_async_le(8);
      } else {
        wait_async_le(0);
      }
      const float* al = As[kc & 1] + arow * APAD;
#pragma unroll
      for (int kk = 0; kk < KC; kk += 4) {
        v2f a, b;
        a.x = al[kk + kh];
        a.y = al[kk + kh + 1];
        b.x = B[(size_t)(k0 + kk + kh) * ROWLEN + c0 + bcol];
        b.y = B[(size_t)(k0 + kk + kh + 1) * ROWLEN + c0 + bcol];
        acc = wmma4(a, b, acc);
      }
    }
  }
#pragma unroll
  for (int v = 0; v < 8; ++v) {
    int row = v + lhalf * 8;
    cm[(size_t)(j0 + row) * C_CH + c0 + bcol] = acc[v] * (1.f / N_NODES);
  }
}

// ---------------------------------------------------------------------------
// K5: out[i,c] = sum_k in[i,k] * w[c,k] + bias[c]   (q / k projection, WMMA)
// ---------------------------------------------------------------------------
__global__ void k_proj(const float* __restrict__ in, const float* __restrict__ w,
                       const float* __restrict__ bias, float* __restrict__ out) {
  int it = blockIdx.x >> 3, ct = blockIdx.x & 7;
  int i0 = it * 16, c0 = ct * 16;
  int lane = threadIdx.x & 31;
  int lhalf = lane >> 4;
  int arow = lane & 15, bcol = lane & 15;
  int kh = lhalf * 2;
  v8f acc = {};
  for (int k0 = 0; k0 < C_CH; k0 += 4) {
    v2f a, b;
    a.x = in[(i0 + arow) * C_CH + k0 + kh];
    a.y = in[(i0 + arow) * C_CH + k0 + kh + 1];
    b.x = w[(c0 + bcol) * C_CH + k0 + kh];
    b.y = w[(c0 + bcol) * C_CH + k0 + kh + 1];
    acc = wmma4(a, b, acc);
  }
#pragma unroll
  for (int v = 0; v < 8; ++v) {
    int row = v + lhalf * 8;
    out[(i0 + row) * C_CH + c0 + bcol] = acc[v] + bias[c0 + bcol];
  }
}

// ---------------------------------------------------------------------------
// K6: attention scores (WMMA, K=D=16), batch mask, softmax over j in LDS
// block = 8 waves; blockIdx = (i-tile, head). LDS holds 16 x 1024 scores.
// ---------------------------------------------------------------------------
__global__ void k_attn(const float* __restrict__ Q, const float* __restrict__ Km,
                       const int* __restrict__ batch, float* __restrict__ ATT) {
  int i0 = blockIdx.x * 16;
  int h = blockIdx.y;
  int wv = threadIdx.x >> 5;
  int lane = threadIdx.x & 31;
  int lhalf = lane >> 4;
  int arow = lane & 15, bcol = lane & 15;
  int kh = lhalf * 2;
  __shared__ float s[16 * N_NODES];                 // 64 KB

  for (int jt = wv; jt < N_NODES / 16; jt += 8) {
    int j0 = jt * 16;
    v8f acc = {};
#pragma unroll
    for (int k0 = 0; k0 < D_HEAD; k0 += 4) {
      v2f a, b;
      a.x = Q[(i0 + arow) * C_CH + h * D_HEAD + k0 + kh];
      a.y = Q[(i0 + arow) * C_CH + h * D_HEAD + k0 + kh + 1];
      b.x = Km[(j0 + bcol) * C_CH + h * D_HEAD + k0 + kh];
      b.y = Km[(j0 + bcol) * C_CH + h * D_HEAD + k0 + kh + 1];
      acc = wmma4(a, b, acc);
    }
#pragma unroll
    for (int v = 0; v < 8; ++v) {
      int row = v + lhalf * 8;
      int j = j0 + bcol;
      s[row * N_NODES + j] =
          (batch[i0 + row] == batch[j]) ? acc[v] * 0.25f : -3.0e38f;
    }
  }
  __syncthreads();

  for (int r = wv * 2; r < wv * 2 + 2; ++r) {
    float mx = -3.0e38f;
    for (int j = lane; j < N_NODES; j += 32) mx = fmaxf(mx, s[r * N_NODES + j]);
#pragma unroll
    for (int o = 16; o > 0; o >>= 1) mx = fmaxf(mx, __shfl_xor(mx, o, 32));
    float sum = 0.f;
    for (int j = lane; j < N_NODES; j += 32) {
      float e = __expf(s[r * N_NODES + j] - mx);
      s[r * N_NODES + j] = e;
      sum += e;
    }
#pragma unroll
    for (int o = 16; o > 0; o >>= 1) sum += __shfl_xor(sum, o, 32);
    float inv = (sum > 0.f) ? 1.f / sum : 0.f;
    for (int j = lane; j < N_NODES; j += 32)
      ATT[((size_t)h * N_NODES + (i0 + r)) * N_NODES + j] = s[r * N_NODES + j] * inv;
  }
}

// ---------------------------------------------------------------------------
// K7: per-m GEMM against per-degree weights [L,C,C]:
//     out[i,m,c] = sum_k in[i,m,k] * wts[l(m)][c,k] (+ bias[c] for m==0)
// ---------------------------------------------------------------------------
__global__ void k_mgemm(const float* __restrict__ in, const float* __restrict__ wts,
                        const float* __restrict__ bias, float* __restrict__ out) {
  int m = blockIdx.y;
  int l = l_of_m(m);
  int it = blockIdx.x >> 3, ct = blockIdx.x & 7;
  int i0 = it * 16, c0 = ct * 16;
  int lane = threadIdx.x & 31;
  int lhalf = lane >> 4;
  int arow = lane & 15, bcol = lane & 15;
  int kh = lhalf * 2;
  const float* W = wts + (size_t)l * C_CH * C_CH;
  v8f acc = {};
  for (int k0 = 0; k0 < C_CH; k0 += 4) {
    v2f a, b;
    a.x = in[(size_t)(i0 + arow) * ROWLEN + m * C_CH + k0 + kh];
    a.y = in[(size_t)(i0 + arow) * ROWLEN + m * C_CH + k0 + kh + 1];
    b.x = W[(c0 + bcol) * C_CH + k0 + kh];
    b.y = W[(c0 + bcol) * C_CH + k0 + kh + 1];
    acc = wmma4(a, b, acc);
  }
  float bv = (bias != nullptr && m == 0) ? bias[c0 + bcol] : 0.f;
#pragma unroll
  for (int v = 0; v < 8; ++v) {
    int row = v + lhalf * 8;
    out[(size_t)(i0 + row) * ROWLEN + m * C_CH + c0 + bcol] = acc[v] + bv;
  }
}

// ---------------------------------------------------------------------------
// K8: attention apply:  O[i,m,h*16+d] = sum_j ATT[h,i,j] * V[j,m,h*16+d]
// A (attn rows) async-staged to LDS, double-buffered; K = N stepped by 4.
// ---------------------------------------------------------------------------
__global__ void k_apply(const float* __restrict__ ATT, const float* __restrict__ V,
                        float* __restrict__ O) {
  int i0 = blockIdx.x * 16;
  int m = blockIdx.y;
  int h = blockIdx.z;
  int lane = threadIdx.x & 31;
  int lhalf = lane >> 4;
  int arow = lane & 15, bcol = lane & 15;
  int kh = lhalf * 2;
  __shared__ __align__(16) float As[2][16 * APAD];
  const float* A = ATT + (size_t)h * N_NODES * N_NODES + (size_t)i0 * N_NODES;
  stage_a_tile(A, N_NODES, 0, As[0], lane);
  v8f acc = {};
  for (int kc = 0; kc < N_NODES / KC; ++kc) {
    int k0 = kc * KC;
    if (kc + 1 < N_NODES / KC) {
      stage_a_tile(A, N_NODES, k0 + KC, As[(kc + 1) & 1], lane);
      wait_async_le(8);
    } else {
      wait_async_le(0);
    }
    const float* al = As[kc & 1] + arow * APAD;
#pragma unroll
    for (int kk = 0; kk < KC; kk += 4) {
      v2f a, b;
      a.x = al[kk + kh];
      a.y = al[kk + kh + 1];
      b.x = V[(size_t)(k0 + kk + kh) * ROWLEN + m * C_CH + h * D_HEAD + bcol];
      b.y = V[(size_t)(k0 + kk + kh + 1) * ROWLEN + m * C_CH + h * D_HEAD + bcol];
      acc = wmma4(a, b, acc);
    }
  }
#pragma unroll
  for (int v = 0; v < 8; ++v) {
    int row = v + lhalf * 8;
    O[(size_t)(i0 + row) * ROWLEN + m * C_CH + h * D_HEAD + bcol] = acc[v];
  }
}

// ---------------------------------------------------------------------------
// K9: residual + LayerNorm over C per (i,m) row; 128 threads = 4 waves
// ---------------------------------------------------------------------------
__global__ void k_ln(const float* __restrict__ x_emb, const float* __restrict__ OO,
                     const float* __restrict__ ln_g, const float* __restrict__ ln_b,
                     float* __restrict__ out) {
  int row = blockIdx.x;                 // i*9 + m
  int i = row / NM, m = row % NM;
  int l = l_of_m(m);
  int c = threadIdx.x;
  int wv = c >> 5, lane = c & 31;
  size_t base = (size_t)i * ROWLEN + m * C_CH;
  float y = x_emb[base + c] + OO[base + c];
  __shared__ float red[4];
  float v = y;
#pragma unroll
  for (int o = 16; o > 0; o >>= 1) v += __shfl_xor(v, o, 32);
  if (lane == 0) red[wv] = v;
  __syncthreads();
  float mu = (red[0] + red[1] + red[2] + red[3]) * (1.f / C_CH);
  __syncthreads();
  float d = y - mu;
  v = d * d;
#pragma unroll
  for (int o = 16; o > 0; o >>= 1) v += __shfl_xor(v, o, 32);
  if (lane == 0) red[wv] = v;
  __syncthreads();
  float var = (red[0] + red[1] + red[2] + red[3]) * (1.f / C_CH);
  out[base + c] = d * rsqrtf(var + 1e-5f) * ln_g[l * C_CH + c] + ln_b[l * C_CH + c];
}

// ---------------------------------------------------------------------------
extern "C" void kernel_launch(void* const* d_in, const int* in_sizes, int n_in,
                              void* d_out, int out_size, void* d_ws, size_t ws_size,
                              hipStream_t stream) {
  (void)in_sizes; (void)n_in; (void)out_size; (void)ws_size;
  const float* x_emb = (const float*)d_in[0];
  const float* pos   = (const float*)d_in[1];
  const float* q_w   = (const float*)d_in[2];
  const float* q_b   = (const float*)d_in[3];
  const float* k_w   = (const float*)d_in[4];
  const float* k_b   = (const float*)d_in[5];
  const float* v_w   = (const float*)d_in[6];
  const float* v_b0  = (const float*)d_in[7];
  const float* out_w = (const float*)d_in[8];
  const float* ln_g  = (const float*)d_in[9];
  const float* ln_b  = (const float*)d_in[10];
  const int*   batch = (const int*)d_in[11];
  float* out = (float*)d_out;

  const size_t NNs = (size_t)N_NODES * N_NODES;
  float* W   = (float*)d_ws;
  float* SHW = W;                        // 9*N*N floats
  float* ATT = W;                        // aliases SHW: SHW dead before attn
  float* P   = W + 9 * NNs;
  float* Wrow = P;  P += (size_t)N_NODES * NM;
  float* rm   = P;  P += (size_t)N_NODES * C_CH;
  float* cm   = P;  P += (size_t)N_NODES * C_CH;
  float* Qm   = P;  P += (size_t)N_NODES * C_CH;
  float* Km   = P;  P += (size_t)N_NODES * C_CH;
  float* V    = P;  P += (size_t)N_NODES * ROWLEN;
  float* O    = P;  P += (size_t)N_NODES * ROWLEN;
  float* OO   = P;  P += (size_t)N_NODES * ROWLEN;

  k_pair_sh<<<(N_NODES * N_NODES) / 256, 256, 0, stream>>>(pos, batch, SHW);
  k_rowsum<<<(NM * N_NODES + 255) / 256, 256, 0, stream>>>(SHW, Wrow);
  k_rowmean<<<(N_NODES * C_CH) / 256, 256, 0, stream>>>(x_emb, Wrow, rm);
  k_colmean<<<(N_NODES / 16) * (C_CH / 16), 32, 0, stream>>>(SHW, x_emb, cm);
  k_proj<<<(N_NODES / 16) * (C_CH / 16), 32, 0, stream>>>(rm, q_w, q_b, Qm);
  k_proj<<<(N_NODES / 16) * (C_CH / 16), 32, 0, stream>>>(cm, k_w, k_b, Km);
  k_attn<<<dim3(N_NODES / 16, H_HEADS), 256, 0, stream>>>(Qm, Km, batch, ATT);
  k_mgemm<<<dim3((N_NODES / 16) * (C_CH / 16), NM), 32, 0, stream>>>(x_emb, v_w, v_b0, V);
  k_apply<<<dim3(N_NODES / 16, NM, H_HEADS), 32, 0, stream>>>(ATT, V, O);
  k_mgemm<<<dim3((N_NODES / 16) * (C_CH / 16), NM), 32, 0, stream>>>(O, out_w, nullptr, OO);
  k_ln<<<N_NODES * NM, C_CH, 0, stream>>>(x_emb, OO, ln_g, ln_b, out);
}